// MoELayer_67130338836772
// MI455X (gfx1250) — compile-verified
//
#include <hip/hip_runtime.h>
#include <hip/hip_bf16.h>

typedef float v2f __attribute__((ext_vector_type(2)));
typedef float v8f __attribute__((ext_vector_type(8)));

#define NB      4
#define NS      2048
#define NTOK    (NB * NS)      // 8192 tokens
#define DIN     1024
#define DOUT    1024
#define NE      16
#define TOPK    2

// ---------------------------------------------------------------------------
// Kernel 0: repack expert_w[:, :, 0:2] -> wcol[e][j][d] (contiguous in d) and
// expert_b[:, 0:2] -> ebias2[e*2+j]. 128 KB compact working set, fully
// coalesced reads in the per-token dot-product kernel.
// ---------------------------------------------------------------------------
__global__ __launch_bounds__(256) void moe_repack(
    const float* __restrict__ expert_w,   // [E, DIN, DOUT]
    const float* __restrict__ expert_b,   // [E, DOUT]
    float* __restrict__ wcol,             // [E, 2, DIN]
    float* __restrict__ ebias2)           // [E, 2]
{
    int i = blockIdx.x * 256 + threadIdx.x;      // 0 .. E*2*DIN-1
    int d = i & (DIN - 1);
    int j = (i >> 10) & 1;
    int e = i >> 11;
    wcol[i] = expert_w[((size_t)e * DIN + d) * DOUT + j];
    if (i < NE * TOPK) {
        ebias2[i] = expert_b[(size_t)(i >> 1) * DOUT + (i & 1)];
    }
}

// ---------------------------------------------------------------------------
// Kernel 1: gating GEMM via V_WMMA_F32_16X16X4_F32.
// One wave = 16 tokens. C tile = 16 tokens (M) x 16 experts (N), f32 acc.
// Then sigmoid, top-2 routing (16-lane butterfly), prob normalization.
// ---------------------------------------------------------------------------
__global__ __launch_bounds__(128) void moe_gate_wmma(
    const float* __restrict__ x,          // [NTOK, DIN]
    const float* __restrict__ gate_w,     // [DIN, NE]
    const float* __restrict__ gate_b,     // [NE]
    const float* __restrict__ ebias_in,   // [NE] routing bias
    float* __restrict__ out_probs,        // [NTOK, NE]
    float* __restrict__ out_idx,          // [NTOK, 2] (written as float)
    float* __restrict__ wprob,            // [NTOK, 2] normalized top-k probs
    int*   __restrict__ widx)             // [NTOK, 2]
{
    const int lane = threadIdx.x & 31;
    const int waveId = blockIdx.x * 4 + (threadIdx.x >> 5);
    const int t0 = waveId * 16;           // token tile base
    const int nl = lane & 15;             // N index (expert) / M row for A
    const int hi = lane >> 4;             // half-wave select

    // A (16x4 f32): lanes 0-15 -> {K=0,K=1}, lanes 16-31 -> {K=2,K=3}, M = nl
    // B (4x16 f32): lanes 0-15 -> {K=0,K=1}, lanes 16-31 -> {K=2,K=3}, N = nl
    const float* aptr = x + (size_t)(t0 + nl) * DIN + hi * 2;
    const float* bptr = gate_w + (size_t)(hi * 2) * NE + nl;

    v8f c = {};
#pragma unroll 4
    for (int k = 0; k < DIN; k += 4) {
        v2f a = *(const v2f*)(aptr + k);                 // 8B load
        v2f b;
        b.x = bptr[(size_t)k * NE];
        b.y = bptr[(size_t)k * NE + NE];
        c = __builtin_amdgcn_wmma_f32_16x16x4_f32(
            /*neg_a=*/false, a, /*neg_b=*/false, b,
            /*c_mod=*/(short)0, c, /*reuse_a=*/false, /*reuse_b=*/false);
    }

    const float gb = gate_b[nl];
    const float eb = ebias_in[nl];

    // C/D layout: VGPR v, lanes 0-15 -> M=v, lanes 16-31 -> M=v+8; N = nl.
#pragma unroll
    for (int v = 0; v < 8; ++v) {
        const int t = t0 + v + hi * 8;
        const float go = c[v] + gb;                      // gate_output[t][nl]
        const float prob = 1.0f / (1.0f + __expf(-go));  // sigmoid
        const float logit = go + eb;                     // routing logit

        out_probs[(size_t)t * NE + nl] = prob;

        // top-1 over 16 experts (xor masks 1..8 stay inside the 16-lane half)
        float v1 = logit; int i1 = nl;
#pragma unroll
        for (int m = 1; m <= 8; m <<= 1) {
            float ov = __shfl_xor(v1, m, 32);
            int   oi = __shfl_xor(i1, m, 32);
            if (ov > v1 || (ov == v1 && oi < i1)) { v1 = ov; i1 = oi; }
        }
        // top-2: mask out winner, reduce again
        float v2 = (nl == i1) ? -__builtin_inff() : logit; int i2 = nl;
#pragma unroll
        for (int m = 1; m <= 8; m <<= 1) {
            float ov = __shfl_xor(v2, m, 32);
            int   oi = __shfl_xor(i2, m, 32);
            if (ov > v2 || (ov == v2 && oi < i2)) { v2 = ov; i2 = oi; }
        }
        // gather sigmoid probs of the two winners from their owning lanes
        float p1 = __shfl(prob, (hi << 4) + i1, 32);
        float p2 = __shfl(prob, (hi << 4) + i2, 32);

        if (nl == 0) {
            float s = p1 + p2;
            wprob[t * 2 + 0] = p1 / s;
            wprob[t * 2 + 1] = p2 / s;
            widx[t * 2 + 0] = i1;
            widx[t * 2 + 1] = i2;
            out_idx[t * 2 + 0] = (float)i1;
            out_idx[t * 2 + 1] = (float)i2;
        }
    }
}

// ---------------------------------------------------------------------------
// Kernel 2: per-token expert dot products + broadcast store.
// One wave per token. x re-read hits L2 (32 MB << 192 MB L2); wcol is 128 KB.
// The 32 MB float4 broadcast store dominates traffic (HBM-bound).
// ---------------------------------------------------------------------------
__global__ __launch_bounds__(256) void moe_expert_out(
    const float* __restrict__ x,        // [NTOK, DIN]
    const float* __restrict__ wcol,     // [E, 2, DIN]
    const float* __restrict__ ebias2,   // [E, 2]
    const float* __restrict__ wprob,    // [NTOK, 2]
    const int*   __restrict__ widx,     // [NTOK, 2]
    float* __restrict__ out)            // [NTOK, DOUT]
{
    const int t = (blockIdx.x * 256 + threadIdx.x) >> 5;   // token = wave id
    const int lane = threadIdx.x & 31;

    const int e1 = widx[t * 2 + 0];
    const int e2 = widx[t * 2 + 1];
    const float w1 = wprob[t * 2 + 0];
    const float w2 = wprob[t * 2 + 1];

    const float* xr = x + (size_t)t * DIN;
    const float* c1 = wcol + (size_t)(e1 * 2 + 0) * DIN;   // expert_w[e1,:,0]
    const float* c2 = wcol + (size_t)(e2 * 2 + 1) * DIN;   // expert_w[e2,:,1]

    float a1 = 0.0f, a2 = 0.0f;
#pragma unroll 8
    for (int d = lane; d < DIN; d += 32) {
        float xv = xr[d];
        a1 = fmaf(xv, c1[d], a1);
        a2 = fmaf(xv, c2[d], a2);
    }
#pragma unroll
    for (int m = 16; m >= 1; m >>= 1) {
        a1 += __shfl_xor(a1, m, 32);
        a2 += __shfl_xor(a2, m, 32);
    }

    const float g1 = a1 + ebias2[e1 * 2 + 0];
    const float g2 = a2 + ebias2[e2 * 2 + 1];
    const float fs = w1 * g1 + w2 * g2;

    float4 f4 = make_float4(fs, fs, fs, fs);
    float4* op = (float4*)(out + (size_t)t * DOUT);
#pragma unroll
    for (int it = 0; it < DOUT / (32 * 4); ++it)           // 8 x b128 stores
        op[it * 32 + lane] = f4;
}

// ---------------------------------------------------------------------------
extern "C" void kernel_launch(void* const* d_in, const int* in_sizes, int n_in,
                              void* d_out, int out_size, void* d_ws, size_t ws_size,
                              hipStream_t stream) {
    const float* x            = (const float*)d_in[0];
    const float* gate_w       = (const float*)d_in[1];
    const float* gate_b       = (const float*)d_in[2];
    const float* expert_w     = (const float*)d_in[3];
    const float* expert_b     = (const float*)d_in[4];
    const float* expert_bias  = (const float*)d_in[5];

    float* out       = (float*)d_out;
    float* out_final = out;                                    // [NTOK, DOUT]
    float* out_probs = out + (size_t)NTOK * DOUT;              // [NTOK, NE]
    float* out_idx   = out_probs + (size_t)NTOK * NE;          // [NTOK, 2]

    float* ws     = (float*)d_ws;
    float* wcol   = ws;                                        // E*2*DIN
    float* ebias2 = wcol + NE * TOPK * DIN;                    // E*2
    float* wprob  = ebias2 + NE * TOPK;                        // NTOK*2
    int*   widx   = (int*)(wprob + NTOK * TOPK);               // NTOK*2

    // 0) compact expert columns (independent of gating)
    moe_repack<<<(NE * TOPK * DIN) / 256, 256, 0, stream>>>(
        expert_w, expert_b, wcol, ebias2);

    // 1) gating GEMM (WMMA f32) + sigmoid + top-2 routing
    moe_gate_wmma<<<NTOK / (16 * 4), 128, 0, stream>>>(
        x, gate_w, gate_b, expert_bias, out_probs, out_idx, wprob, widx);

    // 2) per-token expert dots + broadcast output
    moe_expert_out<<<NTOK / 8, 256, 0, stream>>>(
        x, wcol, ebias2, wprob, widx, out_final);
}